// GIN_18657337933833
// MI455X (gfx1250) — compile-verified
//
#include <hip/hip_runtime.h>
#include <math.h>

#define NNODES 50000
#define HID    64
#define NLAYER 3
#define NEDGE  800000
#define NGRAPH 500
#define NCLS   2
#define TEMP_INV 2.0f      /* 1/0.5 */
#define NEG_W  1.0f
#define BN_EPS 1e-5f

typedef __attribute__((ext_vector_type(2))) float v2f;
typedef __attribute__((ext_vector_type(8))) float v8f;

// ---------------------------------------------------------------------------
// Edge scatter-add: agg[dst] += x[src]. 16 threads/edge, float4 per thread.
// ---------------------------------------------------------------------------
__global__ __launch_bounds__(256) void gin_scatter_kernel(
    const float* __restrict__ x, const int* __restrict__ src,
    const int* __restrict__ dst, float* __restrict__ agg)
{
    int t = blockIdx.x * 256 + threadIdx.x;
    if (t >= NEDGE * 16) return;
    int e  = t >> 4;
    int c4 = (t & 15) << 2;
    int s = src[e];
    int d = dst[e];
    const float4 v = *(const float4*)(x + s * HID + c4);
    float* p = agg + d * HID + c4;
    atomicAdd(p + 0, v.x);
    atomicAdd(p + 1, v.y);
    atomicAdd(p + 2, v.z);
    atomicAdd(p + 3, v.w);
}

// ---------------------------------------------------------------------------
// Fused GIN layer: h=(1+eps)*x+agg ; relu(h@W1+b1) ; relu(.@W2+b2) ; BN ; relu
// One wave = 16 rows. 4 waves/block. WMMA f32 16x16x4, K=64 in 16 steps.
// A frag (16x4 f32): lane L -> M = L&15, K-pair base = 2*(L>>4); v0=K, v1=K+1
// B frag (4x16 f32): lane L -> N = L&15, same K-pair base
// C/D (16x16 f32):   vgpr v -> M = v + 8*(L>>4), N = L&15
// ---------------------------------------------------------------------------
__global__ __launch_bounds__(128) void gin_mlp_kernel(
    const float* x, const float* __restrict__ agg,
    const float* __restrict__ W1, const float* __restrict__ b1,
    const float* __restrict__ W2, const float* __restrict__ b2,
    const float* __restrict__ gamma, const float* __restrict__ beta,
    const float* __restrict__ mean, const float* __restrict__ var,
    const float* __restrict__ epsv, int layer, float* xout)
{
    __shared__ float sW1[HID * HID];
    __shared__ float sW2[HID * HID];
    __shared__ float sb1[HID], sb2[HID], sscale[HID], sshift[HID];
    __shared__ float sh[4][16 * 65];   // per-wave 16x64 tile, row stride 65

    const int tid  = threadIdx.x;
    const int wave = tid >> 5;
    const int lane = tid & 31;

    const float* W1l = W1 + layer * HID * HID;
    const float* W2l = W2 + layer * HID * HID;
    for (int i = tid; i < HID * HID; i += 128) { sW1[i] = W1l[i]; sW2[i] = W2l[i]; }
    if (tid < HID) {
        sb1[tid] = b1[layer * HID + tid];
        sb2[tid] = b2[layer * HID + tid];
        float sc = gamma[layer * HID + tid] * rsqrtf(var[layer * HID + tid] + BN_EPS);
        sscale[tid] = sc;
        sshift[tid] = beta[layer * HID + tid] - mean[layer * HID + tid] * sc;
    }
    const float epl = 1.0f + epsv[layer];
    __syncthreads();

    const int tile = blockIdx.x * 4 + wave;
    const int r0   = tile * 16;
    if (r0 >= NNODES) return;          // wave-uniform exit; no barriers below

    float* hs = sh[wave];
    // stage h = (1+eps)*x + agg  (coalesced: 32 lanes x 32 iters = 1024 elems)
    for (int i = 0; i < 32; i++) {
        int flat = i * 32 + lane;
        int row = flat >> 6, col = flat & 63;
        int g = (r0 + row) * HID + col;
        hs[row * 65 + col] = epl * x[g] + agg[g];
    }

    const int arow = lane & 15;
    const int grp  = lane >> 4;
    const int ncol = lane & 15;

    v8f acc[4];
    // ---- matmul1: relu(h @ W1 + b1) ----
    for (int ct = 0; ct < 4; ct++) {
        v8f c = {0.f,0.f,0.f,0.f,0.f,0.f,0.f,0.f};
        #pragma unroll
        for (int kk = 0; kk < 16; kk++) {
            const int kb = kk * 4 + grp * 2;
            v2f a; a.x = hs[arow * 65 + kb];          a.y = hs[arow * 65 + kb + 1];
            v2f b; b.x = sW1[kb * HID + ct * 16 + ncol];
                   b.y = sW1[(kb + 1) * HID + ct * 16 + ncol];
            c = __builtin_amdgcn_wmma_f32_16x16x4_f32(false, a, false, b,
                                                      (short)0, c, false, false);
        }
        const float bias = sb1[ct * 16 + ncol];
        #pragma unroll
        for (int v = 0; v < 8; v++) c[v] = fmaxf(c[v] + bias, 0.0f);
        acc[ct] = c;
    }
    // write act1 back into the wave-private tile
    #pragma unroll
    for (int ct = 0; ct < 4; ct++)
        #pragma unroll
        for (int v = 0; v < 8; v++)
            hs[(v + grp * 8) * 65 + ct * 16 + ncol] = acc[ct][v];

    // ---- matmul2 + bias + relu + BN + relu ----
    for (int ct = 0; ct < 4; ct++) {
        v8f c = {0.f,0.f,0.f,0.f,0.f,0.f,0.f,0.f};
        #pragma unroll
        for (int kk = 0; kk < 16; kk++) {
            const int kb = kk * 4 + grp * 2;
            v2f a; a.x = hs[arow * 65 + kb];          a.y = hs[arow * 65 + kb + 1];
            v2f b; b.x = sW2[kb * HID + ct * 16 + ncol];
                   b.y = sW2[(kb + 1) * HID + ct * 16 + ncol];
            c = __builtin_amdgcn_wmma_f32_16x16x4_f32(false, a, false, b,
                                                      (short)0, c, false, false);
        }
        const int col = ct * 16 + ncol;
        const float bias = sb2[col], scl = sscale[col], sft = sshift[col];
        #pragma unroll
        for (int v = 0; v < 8; v++) {
            float t = fmaxf(c[v] + bias, 0.0f);   // relu(mlp2)
            t = t * scl + sft;                    // BN (eval)
            t = fmaxf(t, 0.0f);                   // outer relu
            xout[(r0 + v + grp * 8) * HID + col] = t;
        }
    }
}

// ---------------------------------------------------------------------------
// global_add_pool: pooled[batch[i]] += x[i]
// ---------------------------------------------------------------------------
__global__ __launch_bounds__(256) void pool_kernel(
    const float* __restrict__ x, const int* __restrict__ batchv,
    float* __restrict__ pooled)
{
    int t = blockIdx.x * 256 + threadIdx.x;
    if (t >= NNODES * 16) return;
    int i  = t >> 4;
    int c4 = (t & 15) << 2;
    int g  = batchv[i];
    const float4 v = *(const float4*)(x + i * HID + c4);
    float* p = pooled + g * HID + c4;
    atomicAdd(p + 0, v.x);
    atomicAdd(p + 1, v.y);
    atomicAdd(p + 2, v.z);
    atomicAdd(p + 3, v.w);
}

// ---------------------------------------------------------------------------
// L2 normalize rows (B x 64)
// ---------------------------------------------------------------------------
__global__ __launch_bounds__(64) void l2norm_kernel(
    const float* __restrict__ p, float* __restrict__ outn)
{
    __shared__ float red[64];
    const int r = blockIdx.x, tid = threadIdx.x;
    float v = p[r * HID + tid];
    red[tid] = v * v;
    __syncthreads();
    for (int s = 32; s > 0; s >>= 1) {
        if (tid < s) red[tid] += red[tid + s];
        __syncthreads();
    }
    float nrm = fmaxf(sqrtf(red[0]), 1e-12f);
    outn[r * HID + tid] = v / nrm;
}

// ---------------------------------------------------------------------------
// Classification head: concat -> fc1 relu -> fc2 relu -> fc3 -> log_softmax
// ---------------------------------------------------------------------------
__global__ __launch_bounds__(128) void head_kernel(
    const float* __restrict__ scp, const float* __restrict__ fcp,
    const float* __restrict__ W1, const float* __restrict__ b1,
    const float* __restrict__ W2, const float* __restrict__ b2,
    const float* __restrict__ W3, const float* __restrict__ b3,
    float* __restrict__ out3)
{
    __shared__ float z[2 * HID];
    __shared__ float h1[HID];
    __shared__ float h2[HID / 2];
    __shared__ float lg[NCLS];
    const int r = blockIdx.x, tid = threadIdx.x;
    z[tid] = (tid < HID) ? scp[r * HID + tid] : fcp[r * HID + tid - HID];
    __syncthreads();
    if (tid < HID) {
        float a = b1[tid];
        for (int k = 0; k < 2 * HID; k++) a += z[k] * W1[k * HID + tid];
        h1[tid] = fmaxf(a, 0.0f);
    }
    __syncthreads();
    if (tid < HID / 2) {
        float a = b2[tid];
        for (int k = 0; k < HID; k++) a += h1[k] * W2[k * (HID / 2) + tid];
        h2[tid] = fmaxf(a, 0.0f);
    }
    __syncthreads();
    if (tid < NCLS) {
        float a = b3[tid];
        for (int k = 0; k < HID / 2; k++) a += h2[k] * W3[k * NCLS + tid];
        lg[tid] = a;
    }
    __syncthreads();
    if (tid == 0) {
        float m = fmaxf(lg[0], lg[1]);
        float lse = m + logf(expf(lg[0] - m) + expf(lg[1] - m));
        out3[r * NCLS + 0] = lg[0] - lse;
        out3[r * NCLS + 1] = lg[1] - lse;
    }
}

// ---------------------------------------------------------------------------
// Contrastive loss: per anchor row, logits = [cross(all j), self(j!=i), 0 at j==i]
// loss_row = logsumexp(row) - cross(i,i). blockIdx.y: 0 = sc anchor, 1 = fc.
// ---------------------------------------------------------------------------
__global__ __launch_bounds__(128) void loss_kernel(
    const float* __restrict__ scn, const float* __restrict__ fcn,
    float* __restrict__ lossacc)
{
    __shared__ float anc[HID];
    __shared__ float lg[2 * NGRAPH];
    __shared__ float red[128];
    const int i = blockIdx.x, which = blockIdx.y, tid = threadIdx.x;
    const float* A  = which ? fcn : scn;
    const float* Cr = which ? scn : fcn;
    if (tid < HID) anc[tid] = A[i * HID + tid];
    __syncthreads();
    for (int j = tid; j < NGRAPH; j += 128) {
        float d1 = 0.0f, d2 = 0.0f;
        for (int k = 0; k < HID; k++) {
            float a = anc[k];
            d1 += a * Cr[j * HID + k];
            d2 += a * A [j * HID + k];
        }
        lg[j]          = d1 * TEMP_INV;
        lg[NGRAPH + j] = (j == i) ? 0.0f : d2 * TEMP_INV * NEG_W;  // pos_mask
    }
    __syncthreads();
    float m = -3.4e38f;
    for (int j = tid; j < 2 * NGRAPH; j += 128) m = fmaxf(m, lg[j]);
    red[tid] = m;
    __syncthreads();
    for (int s = 64; s > 0; s >>= 1) {
        if (tid < s) red[tid] = fmaxf(red[tid], red[tid + s]);
        __syncthreads();
    }
    m = red[0];
    __syncthreads();
    float sum = 0.0f;
    for (int j = tid; j < 2 * NGRAPH; j += 128) sum += expf(lg[j] - m);
    red[tid] = sum;
    __syncthreads();
    for (int s = 64; s > 0; s >>= 1) {
        if (tid < s) red[tid] += red[tid + s];
        __syncthreads();
    }
    if (tid == 0) {
        float lse = m + logf(red[0]);
        atomicAdd(lossacc, lse - lg[i]);
    }
}

__global__ __launch_bounds__(256) void final_kernel(
    const float* __restrict__ out3, const float* __restrict__ lossacc,
    float* __restrict__ out)
{
    int t = blockIdx.x * 256 + threadIdx.x;
    if (t < NGRAPH * NCLS)
        out[t] = out3[t] + lossacc[0] * (1.0f / (2.0f * NGRAPH));
}

// ---------------------------------------------------------------------------
extern "C" void kernel_launch(void* const* d_in, const int* in_sizes, int n_in,
                              void* d_out, int out_size, void* d_ws, size_t ws_size,
                              hipStream_t stream)
{
    (void)in_sizes; (void)n_in; (void)out_size; (void)ws_size;
    const float* sc_x = (const float*)d_in[0];
    const float* fc_x = (const float*)d_in[1];
    const float* fc1_W = (const float*)d_in[20];
    const float* fc1_b = (const float*)d_in[21];
    const float* fc2_W = (const float*)d_in[22];
    const float* fc2_b = (const float*)d_in[23];
    const float* fc3_W = (const float*)d_in[24];
    const float* fc3_b = (const float*)d_in[25];
    const int* sc_ei = (const int*)d_in[26];
    const int* fc_ei = (const int*)d_in[27];
    const int* batchv = (const int*)d_in[28];

    float* ws      = (float*)d_ws;
    float* xbuf    = ws;                          // N*64
    float* agg     = xbuf + NNODES * HID;         // N*64
    float* sc_p    = agg + NNODES * HID;          // B*64
    float* fc_p    = sc_p + NGRAPH * HID;
    float* sc_nn   = fc_p + NGRAPH * HID;
    float* fc_nn   = sc_nn + NGRAPH * HID;
    float* out3    = fc_nn + NGRAPH * HID;        // B*2
    float* lossacc = out3 + NGRAPH * NCLS;        // 1

    const int ntiles = (NNODES + 15) / 16;                // 3125
    const int mlp_blocks = (ntiles + 3) / 4;              // 782

    for (int br = 0; br < 2; br++) {
        const float* x0 = br ? fc_x : sc_x;
        const int*   ei = br ? fc_ei : sc_ei;
        const int base = 2 + br * 9;
        const float* W1    = (const float*)d_in[base + 0];
        const float* b1    = (const float*)d_in[base + 1];
        const float* W2    = (const float*)d_in[base + 2];
        const float* b2    = (const float*)d_in[base + 3];
        const float* gamma = (const float*)d_in[base + 4];
        const float* beta  = (const float*)d_in[base + 5];
        const float* meanp = (const float*)d_in[base + 6];
        const float* varp  = (const float*)d_in[base + 7];
        const float* epsv  = (const float*)d_in[base + 8];

        hipMemcpyAsync(xbuf, x0, (size_t)NNODES * HID * sizeof(float),
                       hipMemcpyDeviceToDevice, stream);
        for (int l = 0; l < NLAYER; l++) {
            hipMemsetAsync(agg, 0, (size_t)NNODES * HID * sizeof(float), stream);
            gin_scatter_kernel<<<(NEDGE * 16) / 256, 256, 0, stream>>>(
                xbuf, ei, ei + NEDGE, agg);
            gin_mlp_kernel<<<mlp_blocks, 128, 0, stream>>>(
                xbuf, agg, W1, b1, W2, b2, gamma, beta, meanp, varp, epsv, l, xbuf);
        }
        float* pooled = br ? fc_p : sc_p;
        hipMemsetAsync(pooled, 0, (size_t)NGRAPH * HID * sizeof(float), stream);
        pool_kernel<<<(NNODES * 16) / 256, 256, 0, stream>>>(xbuf, batchv, pooled);
    }

    head_kernel<<<NGRAPH, 128, 0, stream>>>(sc_p, fc_p, fc1_W, fc1_b,
                                            fc2_W, fc2_b, fc3_W, fc3_b, out3);
    l2norm_kernel<<<NGRAPH, 64, 0, stream>>>(sc_p, sc_nn);
    l2norm_kernel<<<NGRAPH, 64, 0, stream>>>(fc_p, fc_nn);
    hipMemsetAsync(lossacc, 0, sizeof(float), stream);
    loss_kernel<<<dim3(NGRAPH, 2), 128, 0, stream>>>(sc_nn, fc_nn, lossacc);
    final_kernel<<<(NGRAPH * NCLS + 255) / 256, 256, 0, stream>>>(
        out3, lossacc, (float*)d_out);
}